// RelGraphConv_52458730553706
// MI455X (gfx1250) — compile-verified
//
#include <hip/hip_runtime.h>
#include <hip/hip_bf16.h>

typedef float v2f __attribute__((ext_vector_type(2)));
typedef float v4f __attribute__((ext_vector_type(4)));
typedef float v8f __attribute__((ext_vector_type(8)));

#define NN 100000
#define EE 1600000
#define RR 8
#define DD 128
#define KBIG 1152   // 128 skip + 8*128 relations

// ---------------------------------------------------------------------------
// 0) zero agg [N*1024] + deg [N] (contiguous region, N*1025 floats, /4 exact)
// ---------------------------------------------------------------------------
__global__ void zero_ws_kernel(float* ws) {
    size_t i = (size_t)blockIdx.x * blockDim.x + threadIdx.x;
    size_t nvec = (size_t)NN * 1025 / 4;
    if (i < nvec) ((v4f*)ws)[i] = (v4f){0.f, 0.f, 0.f, 0.f};
}

// ---------------------------------------------------------------------------
// 1) in-degree via hardware fp32 atomics
// ---------------------------------------------------------------------------
__global__ void degree_kernel(const int* __restrict__ dst, float* __restrict__ deg) {
    int e = blockIdx.x * blockDim.x + threadIdx.x;
    if (e < EE) unsafeAtomicAdd(&deg[dst[e]], 1.0f);
}

// ---------------------------------------------------------------------------
// 2) deg -> inv_deg in place
// ---------------------------------------------------------------------------
__global__ void invdeg_kernel(float* deg) {
    int v = blockIdx.x * blockDim.x + threadIdx.x;
    if (v < NN) {
        float d = deg[v];
        deg[v] = (d > 0.f) ? (1.0f / d) : 0.f;
    }
}

// ---------------------------------------------------------------------------
// 3) edge scatter: one wave32 per edge.
//    lane holds cols [4*lane, 4*lane+4) of x[src]; 8 relation weights are
//    loaded by lanes 0..7 and broadcast with __shfl. agg[v] layout: r*128+c.
// ---------------------------------------------------------------------------
__global__ void scatter_kernel(const float* __restrict__ nf,
                               const float* __restrict__ ew,
                               const int* __restrict__ src,
                               const int* __restrict__ dst,
                               float* __restrict__ agg) {
    const int lane = threadIdx.x & 31;
    const int wv   = threadIdx.x >> 5;
    const long e   = (long)blockIdx.x * 8 + wv;   // blockDim = 256 -> 8 edges
    if (e >= EE) return;

    const int s = src[e];
    const int d = dst[e];

    v4f x = *(const v4f*)(nf + (size_t)s * DD + lane * 4);
    float wl = (lane < RR) ? ew[(size_t)lane * EE + e] : 0.f;

    float* base = agg + (size_t)d * (RR * DD) + lane * 4;
#pragma unroll
    for (int r = 0; r < RR; ++r) {
        float w = __shfl(wl, r, 32);
        float* p = base + r * DD;
        unsafeAtomicAdd(p + 0, x.x * w);
        unsafeAtomicAdd(p + 1, x.y * w);
        unsafeAtomicAdd(p + 2, x.z * w);
        unsafeAtomicAdd(p + 3, x.w * w);
    }
}

// ---------------------------------------------------------------------------
// 4) fused GEMM: out[v,:] = b + X_big[v,:] @ W_big  (K = 1152)
//    block = 128 threads (4 waves), M-tile = 16 rows, full 128 cols.
//    Each wave owns 32 cols (2 x 16x16 WMMA accumulators).
//    K chunked by 128 through LDS; inv_deg folded into A-tile load.
// ---------------------------------------------------------------------------
#define AS_STRIDE 132   // 16 x 132 f32  (banks: 4*row step, conflict-free)
#define BS_STRIDE 136   // 128 x 136 f32 (half-wave row offset lands +16 banks)
#define KC 128

__global__ void __launch_bounds__(128)
gemm_wmma_kernel(const float* __restrict__ nf,
                 const float* __restrict__ skipW,
                 const float* __restrict__ relW,
                 const float* __restrict__ skipB,
                 const float* __restrict__ agg,
                 const float* __restrict__ invdeg,
                 float* __restrict__ out) {
    __shared__ float As[16 * AS_STRIDE];
    __shared__ float Bs[KC * BS_STRIDE];

    const int tid  = threadIdx.x;          // 0..127
    const int wave = tid >> 5;             // 0..3
    const int lane = tid & 31;
    const int half = lane >> 4;            // 0: K pair 0/1, 1: K pair 2/3
    const int lrow = lane & 15;
    const int m0   = blockIdx.x * 16;      // 6250 * 16 == 100000 exactly
    const int wbase = wave * 32;

    v8f acc0 = {};
    v8f acc1 = {};

    for (int c = 0; c < KBIG / KC; ++c) {
        const int k0 = c * KC;

        // ---- stage W_big chunk: 128 rows x 128 cols -> Bs -----------------
#pragma unroll
        for (int it = 0; it < 32; ++it) {
            int idx = it * 128 + tid;            // 0..4095 float4 slots
            int row = idx >> 5;                  // 0..127
            int c4  = (idx & 31) << 2;           // 0..124
            int gk  = k0 + row;
            const float* sp = (gk < DD) ? (skipW + (size_t)gk * DD + c4)
                                        : (relW + (size_t)(gk - DD) * DD + c4);
            *(v4f*)&Bs[row * BS_STRIDE + c4] = *(const v4f*)sp;
        }

        // ---- stage X_big tile: 16 rows x 128 cols -> As -------------------
#pragma unroll
        for (int it = 0; it < 4; ++it) {
            int idx = it * 128 + tid;            // 0..511 float4 slots
            int row = idx >> 5;                  // 0..15
            int c4  = (idx & 31) << 2;
            int v   = m0 + row;
            v4f a;
            if (c == 0) {
                a = *(const v4f*)(nf + (size_t)v * DD + c4);
            } else {
                a = *(const v4f*)(agg + (size_t)v * (RR * DD) + (k0 - DD) + c4);
                float s = invdeg[v];
                a.x *= s; a.y *= s; a.z *= s; a.w *= s;
            }
            *(v4f*)&As[row * AS_STRIDE + c4] = a;
        }

        // prefetch next W chunk while we compute this one
        if (c + 1 < KBIG / KC)
            __builtin_prefetch(relW + (size_t)(k0 + KC - DD) * DD + tid * 4, 0, 1);

        __syncthreads();

        // ---- 32 k-steps of v_wmma_f32_16x16x4_f32 per N-tile --------------
#pragma unroll
        for (int kk = 0; kk < KC; kk += 4) {
            const int ak = kk + 2 * half;
            v2f a;
            a.x = As[lrow * AS_STRIDE + ak];
            a.y = As[lrow * AS_STRIDE + ak + 1];

            const int col0 = wbase + lrow;
            const int col1 = wbase + 16 + lrow;
            v2f b0, b1;
            b0.x = Bs[ak * BS_STRIDE + col0];
            b0.y = Bs[(ak + 1) * BS_STRIDE + col0];
            b1.x = Bs[ak * BS_STRIDE + col1];
            b1.y = Bs[(ak + 1) * BS_STRIDE + col1];

            acc0 = __builtin_amdgcn_wmma_f32_16x16x4_f32(
                false, a, false, b0, (short)0, acc0, false, false);
            acc1 = __builtin_amdgcn_wmma_f32_16x16x4_f32(
                false, a, false, b1, (short)0, acc1, false, false);
        }
        __syncthreads();
    }

    // ---- epilogue: C layout VGPR i -> row (half*8 + i), col = lrow ---------
    {
        const int col0 = wbase + lrow;
        const int col1 = wbase + 16 + lrow;
        const float bias0 = skipB[col0];
        const float bias1 = skipB[col1];
        const int rbase = half * 8;
#pragma unroll
        for (int i = 0; i < 8; ++i) {
            int row = m0 + rbase + i;
            out[(size_t)row * DD + col0] = acc0[i] + bias0;
            out[(size_t)row * DD + col1] = acc1[i] + bias1;
        }
    }
}

// ---------------------------------------------------------------------------
// launch
// ---------------------------------------------------------------------------
extern "C" void kernel_launch(void* const* d_in, const int* in_sizes, int n_in,
                              void* d_out, int out_size, void* d_ws, size_t ws_size,
                              hipStream_t stream) {
    const float* nf    = (const float*)d_in[0];   // [N,128]
    const float* ew    = (const float*)d_in[1];   // [R,E]
    const float* relW  = (const float*)d_in[2];   // [R,128,128]
    const float* skipW = (const float*)d_in[3];   // [128,128]
    const float* skipB = (const float*)d_in[4];   // [128]
    const int*   src   = (const int*)d_in[5];     // [E]
    const int*   dst   = (const int*)d_in[6];     // [E]
    float* out = (float*)d_out;                   // [N,128]

    float* agg = (float*)d_ws;                          // N*1024 f32
    float* deg = agg + (size_t)NN * (RR * DD);          // N f32 (-> inv_deg)

    {   // zero agg + deg
        size_t nvec = (size_t)NN * 1025 / 4;
        int blocks = (int)((nvec + 255) / 256);
        zero_ws_kernel<<<blocks, 256, 0, stream>>>((float*)d_ws);
    }
    degree_kernel<<<(EE + 255) / 256, 256, 0, stream>>>(dst, deg);
    invdeg_kernel<<<(NN + 255) / 256, 256, 0, stream>>>(deg);
    scatter_kernel<<<EE / 8, 256, 0, stream>>>(nf, ew, src, dst, agg);
    gemm_wmma_kernel<<<NN / 16, 128, 0, stream>>>(nf, skipW, relW, skipB, agg, deg, out);
}